// GCN_19816979104580
// MI455X (gfx1250) — compile-verified
//
#include <hip/hip_runtime.h>
#include <hip/hip_bf16.h>

// ---------------------------------------------------------------------------
// LightGCN propagation on gfx1250.
// Strategy: build CSR once (histogram + scan + scatter), then each wave owns
// an aligned 16-row output block and accumulates it with chained
// V_WMMA_F32_16X16X4_F32 (A = per-4-edge val/selector matrix, B = gathered x
// rows, C = f32 accumulator). No atomics in the hot loop; everything stays in
// the 192MB L2. Loop control is forced into SGPRs (readfirstlane) so EXEC is
// provably all-ones at every WMMA.
// ---------------------------------------------------------------------------

#define GCN_NU 100000
#define GCN_NI 200000
#define GCN_NN 300000            // total nodes
#define GCN_NE 3200000           // edges
#define GCN_DIM 64
#define GCN_NBLK (GCN_NN / 16)   // 18750 aligned 16-row blocks

typedef float v2f __attribute__((ext_vector_type(2)));
typedef float v8f __attribute__((ext_vector_type(8)));

// ---- init: x = concat(user,item); acc(out) = same ------------------------
__global__ void gcn_init_kernel(const float4* __restrict__ user4,
                                const float4* __restrict__ item4,
                                float4* __restrict__ x4,
                                float4* __restrict__ out4) {
    const int n4 = GCN_NN * GCN_DIM / 4;          // 4.8M float4
    const int nu4 = GCN_NU * GCN_DIM / 4;         // 1.6M float4
    int i = blockIdx.x * blockDim.x + threadIdx.x;
    if (i >= n4) return;
    float4 v = (i < nu4) ? user4[i] : item4[i - nu4];
    x4[i] = v;
    out4[i] = v;
}

// ---- zero an int array ----------------------------------------------------
__global__ void gcn_zero_i32_kernel(int* __restrict__ p, int n) {
    int i = blockIdx.x * blockDim.x + threadIdx.x;
    if (i < n) p[i] = 0;
}

// ---- histogram of row ids -------------------------------------------------
__global__ void gcn_hist_kernel(const int* __restrict__ rows,
                                int* __restrict__ cnt) {
    int e = blockIdx.x * blockDim.x + threadIdx.x;
    if (e >= GCN_NE) return;
    atomicAdd(&cnt[rows[e]], 1);
}

// ---- exclusive scan, 1024 elems / block -----------------------------------
__global__ void gcn_scan1_kernel(const int* __restrict__ cnt,
                                 int* __restrict__ rowPtr,
                                 int* __restrict__ blockSums) {
    __shared__ int sh[1024];
    int tid = threadIdx.x;
    int i = blockIdx.x * 1024 + tid;
    int v = (i < GCN_NN) ? cnt[i] : 0;
    sh[tid] = v;
    __syncthreads();
    for (int off = 1; off < 1024; off <<= 1) {
        int t = (tid >= off) ? sh[tid - off] : 0;
        __syncthreads();
        sh[tid] += t;
        __syncthreads();
    }
    int incl = sh[tid];
    if (i < GCN_NN) rowPtr[i] = incl - v;         // exclusive
    if (tid == 1023) blockSums[blockIdx.x] = incl;
}

__global__ void gcn_scan2_kernel(int* __restrict__ blockSums, int n) {
    __shared__ int sh[1024];
    int tid = threadIdx.x;
    int v = (tid < n) ? blockSums[tid] : 0;
    sh[tid] = v;
    __syncthreads();
    for (int off = 1; off < 1024; off <<= 1) {
        int t = (tid >= off) ? sh[tid - off] : 0;
        __syncthreads();
        sh[tid] += t;
        __syncthreads();
    }
    if (tid < n) blockSums[tid] = sh[tid] - v;    // exclusive
}

// ---- add back block sums; also copy to cursor array -----------------------
__global__ void gcn_scan3_kernel(int* __restrict__ rowPtr,
                                 const int* __restrict__ blockSums,
                                 int* __restrict__ cursors) {
    int i = blockIdx.x * blockDim.x + threadIdx.x;
    if (i >= GCN_NN) return;
    int v = rowPtr[i] + blockSums[i >> 10];
    rowPtr[i] = v;
    cursors[i] = v;
}

// ---- scatter edges into CSR order; pack (col | rlocal<<20, valbits) -------
__global__ void gcn_scatter_kernel(const int* __restrict__ rows,
                                   const int* __restrict__ cols,
                                   const float* __restrict__ vals,
                                   int* __restrict__ cursors,
                                   int2* __restrict__ edgeCV) {
    int e = blockIdx.x * blockDim.x + threadIdx.x;
    if (e >= GCN_NE) return;
    int r = rows[e];
    int pos = atomicAdd(&cursors[r], 1);
    int2 cv;
    cv.x = (int)(((unsigned)cols[e]) | (((unsigned)(r & 15)) << 20));
    cv.y = __float_as_int(vals[e]);
    edgeCV[pos] = cv;
}

// ---- WMMA SpMM: one wave per 16-row block, fused acc += y -----------------
__global__ void __launch_bounds__(128)
gcn_spmm_wmma_kernel(const float* __restrict__ x,
                     float* __restrict__ y,
                     float* __restrict__ acc,
                     const int2* __restrict__ edgeCV,
                     const int* __restrict__ rowPtr) {
    // Force wave-uniform values into SGPRs: scalar loop control, full EXEC
    // at every WMMA, scalar rowPtr loads.
    const int blk = __builtin_amdgcn_readfirstlane(
        (int)(blockIdx.x * 4 + (threadIdx.x >> 5)));
    if (blk >= GCN_NBLK) return;                  // scalar branch

    const int s    = __builtin_amdgcn_readfirstlane(rowPtr[blk * 16]);
    const int eEnd = __builtin_amdgcn_readfirstlane(
        (blk == GCN_NBLK - 1) ? GCN_NE : rowPtr[blk * 16 + 16]);

    const int  lane = threadIdx.x & 31;
    const int  m  = lane & 15;                    // local row (A) / col N (B,C)
    const bool hi = lane >= 16;
    const int  k0 = hi ? 2 : 0;                   // K indices held by this lane
    const int  k1 = k0 + 1;

    v8f c0 = {}, c1 = {}, c2 = {}, c3 = {};

#pragma unroll 2
    for (int base = s; base < eEnd; base += 4) {
        // prefetch the streamed edge array ~1KB ahead (global_prefetch_b8)
        __builtin_prefetch((const char*)(edgeCV + base) + 1024, 0, 0);

        int i0 = base + k0, i1 = base + k1;
        int j0 = (i0 < eEnd) ? i0 : (eEnd - 1);   // clamp for safe load
        int j1 = (i1 < eEnd) ? i1 : (eEnd - 1);
        int2 cv0 = edgeCV[j0];
        int2 cv1 = edgeCV[j1];
        unsigned p0 = (unsigned)cv0.x, p1 = (unsigned)cv1.x;
        int col0 = (int)(p0 & 0xFFFFFu), rl0 = (int)(p0 >> 20);
        int col1 = (int)(p1 & 0xFFFFFu), rl1 = (int)(p1 >> 20);

        // A[m][k]: val if edge k hits local row m (zero-padded past segment)
        v2f a;
        a.x = (i0 < eEnd && rl0 == m) ? __int_as_float(cv0.y) : 0.0f;
        a.y = (i1 < eEnd && rl1 == m) ? __int_as_float(cv1.y) : 0.0f;

        const float* xr0 = x + (size_t)col0 * GCN_DIM;
        const float* xr1 = x + (size_t)col1 * GCN_DIM;

        // B[k][n] per 16-dim tile; lanes 0-15 read 64B contiguous per tile
        v2f b0, b1, b2, b3;
        b0.x = xr0[m];      b0.y = xr1[m];
        b1.x = xr0[16 + m]; b1.y = xr1[16 + m];
        b2.x = xr0[32 + m]; b2.y = xr1[32 + m];
        b3.x = xr0[48 + m]; b3.y = xr1[48 + m];

        c0 = __builtin_amdgcn_wmma_f32_16x16x4_f32(false, a, false, b0,
                                                   (short)0, c0, false, false);
        c1 = __builtin_amdgcn_wmma_f32_16x16x4_f32(false, a, false, b1,
                                                   (short)0, c1, false, false);
        c2 = __builtin_amdgcn_wmma_f32_16x16x4_f32(false, a, false, b2,
                                                   (short)0, c2, false, false);
        c3 = __builtin_amdgcn_wmma_f32_16x16x4_f32(false, a, false, b3,
                                                   (short)0, c3, false, false);
    }

    // C/D layout: vgpr r -> row M = r (lanes 0-15) / r+8 (lanes 16-31), N = m
    const int rowBase = blk * 16 + (hi ? 8 : 0);
#pragma unroll
    for (int r = 0; r < 8; ++r) {
        size_t o = (size_t)(rowBase + r) * GCN_DIM + m;
        float t0 = c0[r], t1 = c1[r], t2 = c2[r], t3 = c3[r];
        y[o]        = t0;  acc[o]      += t0;     // exclusive row ownership:
        y[o + 16]   = t1;  acc[o + 16] += t1;     // plain RMW, no atomics
        y[o + 32]   = t2;  acc[o + 32] += t2;
        y[o + 48]   = t3;  acc[o + 48] += t3;
    }
}

// ---- final scale by 1/(n_layers+1) ---------------------------------------
__global__ void gcn_scale_kernel(float4* __restrict__ out4) {
    const int n4 = GCN_NN * GCN_DIM / 4;
    int i = blockIdx.x * blockDim.x + threadIdx.x;
    if (i >= n4) return;
    float4 v = out4[i];
    v.x *= 0.25f; v.y *= 0.25f; v.z *= 0.25f; v.w *= 0.25f;
    out4[i] = v;
}

extern "C" void kernel_launch(void* const* d_in, const int* in_sizes, int n_in,
                              void* d_out, int out_size, void* d_ws, size_t ws_size,
                              hipStream_t stream) {
    const float* user = (const float*)d_in[0];
    const float* item = (const float*)d_in[1];
    const int*   rows = (const int*)d_in[2];
    const int*   cols = (const int*)d_in[3];
    const float* vals = (const float*)d_in[4];
    float* out = (float*)d_out;
    (void)in_sizes; (void)n_in; (void)out_size; (void)ws_size;

    // workspace layout
    const size_t SZ_X = (size_t)GCN_NN * GCN_DIM * sizeof(float); // 76.8 MB
    char* base = (char*)d_ws;
    float* xbuf   = (float*)(base);
    float* ybuf   = (float*)(base + SZ_X);
    int2*  edgeCV = (int2*)(base + 2 * SZ_X);
    int*   rowPtr = (int*)(base + 2 * SZ_X + (size_t)GCN_NE * sizeof(int2));
    int*   cnt    = rowPtr + GCN_NN;      // doubles as histogram then cursors
    int*   bsums  = cnt + GCN_NN;

    const int scanBlocks = (GCN_NN + 1023) / 1024;                // 293

    // 1) x = concat(user,item); acc = x
    gcn_init_kernel<<<(GCN_NN * GCN_DIM / 4 + 255) / 256, 256, 0, stream>>>(
        (const float4*)user, (const float4*)item, (float4*)xbuf, (float4*)out);

    // 2) CSR build (once; graph constant across layers)
    gcn_zero_i32_kernel<<<(GCN_NN + 255) / 256, 256, 0, stream>>>(cnt, GCN_NN);
    gcn_hist_kernel<<<(GCN_NE + 255) / 256, 256, 0, stream>>>(rows, cnt);
    gcn_scan1_kernel<<<scanBlocks, 1024, 0, stream>>>(cnt, rowPtr, bsums);
    gcn_scan2_kernel<<<1, 1024, 0, stream>>>(bsums, scanBlocks);
    gcn_scan3_kernel<<<(GCN_NN + 255) / 256, 256, 0, stream>>>(rowPtr, bsums, cnt);
    gcn_scatter_kernel<<<(GCN_NE + 255) / 256, 256, 0, stream>>>(
        rows, cols, vals, cnt, edgeCV);

    // 3) three propagation layers, ping-pong x<->y, fused acc +=
    float* src = xbuf;
    float* dst = ybuf;
    const int spmmGrid = (GCN_NBLK + 3) / 4;                       // 4 waves/WG
    for (int l = 0; l < 3; ++l) {
        gcn_spmm_wmma_kernel<<<spmmGrid, 128, 0, stream>>>(
            src, dst, out, edgeCV, rowPtr);
        float* t = src; src = dst; dst = t;
    }

    // 4) out /= 4
    gcn_scale_kernel<<<(GCN_NN * GCN_DIM / 4 + 255) / 256, 256, 0, stream>>>(
        (float4*)out);
}